// StudentSpecificFPFilter_88115549045401
// MI455X (gfx1250) — compile-verified
//
#include <hip/hip_runtime.h>
#include <hip/hip_bf16.h>
#include <stdint.h>

// Problem constants (fixed by the reference setup_inputs()).
#define B_  16
#define H_  1024
#define L_  4096
#define N_  (H_ + L_)     // 5120 concatenated student boxes per batch
#define T_  2048          // teacher boxes per batch
#define C_  10            // classes
#define D_  9             // roi feature dim
#define TQ_ 512           // teachers per quarter-block (grid.y splits T_)
#define NQ_ (T_ / TQ_)    // 4 quarters
#define IOU_T 0.3f
#define PI_ 3.14159265358979323846f

typedef int v4i_ __attribute__((ext_vector_type(4)));

// ---------------------------------------------------------------------------
// CDNA5 async global->LDS copy (ASYNCcnt path), probe-verified signature:
// (v4i addrspace(1)* gsrc, v4i addrspace(3)* ldsdst, imm offset, imm cpol).
// ---------------------------------------------------------------------------
__device__ __forceinline__ void async_cp16(void* lds_dst, const void* g_src) {
#if defined(__HIP_DEVICE_COMPILE__)
#if __has_builtin(__builtin_amdgcn_global_load_async_to_lds_b128)
  __builtin_amdgcn_global_load_async_to_lds_b128(
      (__attribute__((address_space(1))) v4i_*)(g_src),
      (__attribute__((address_space(3))) v4i_*)(lds_dst),
      /*offset=*/0, /*cpol=*/0);
#else
  unsigned int loff = (unsigned int)(uintptr_t)
      (__attribute__((address_space(3))) void*)(lds_dst);
  asm volatile("global_load_async_to_lds_b128 %0, %1, off"
               :
               : "v"(loff), "v"(g_src)
               : "memory");
#endif
#else
  *(float4*)lds_dst = *(const float4*)g_src;   // host pass stub
#endif
}

__device__ __forceinline__ void async_wait_all() {
#if defined(__HIP_DEVICE_COMPILE__)
#if __has_builtin(__builtin_amdgcn_s_wait_asynccnt)
  __builtin_amdgcn_s_wait_asynccnt(0);
#else
  asm volatile("s_wait_asynccnt 0" ::: "memory");
#endif
#endif
}

// nearest_bev: if |limit_period(yaw)| > pi/4 swap dx/dy; return xyxy + area.
__device__ __forceinline__ void nearest_bev(float x, float y, float dx, float dy,
                                            float yaw, float4* box, float* area) {
  float normed = fabsf(yaw - floorf(yaw * (1.0f / PI_) + 0.5f) * PI_);
  bool  swap   = normed > (PI_ * 0.25f);
  float w = swap ? dy : dx;
  float h = swap ? dx : dy;
  *box  = make_float4(x - 0.5f * w, y - 0.5f * h, x + 0.5f * w, y + 0.5f * h);
  *area = w * h;
}

// Workspace layout: u64 keys[160] | u32 hitbits[B_*N_/32] (2560 words).
#define KEYS_WORDS_   (B_ * C_ * 2)           // 160 u64 = 320 u32
#define HITS_WORDS_   (B_ * N_ / 32)          // 2560 u32
#define WS_ZERO_WORDS (KEYS_WORDS_ + HITS_WORDS_)

// ---------------------------------------------------------------------------
// Kernel 0: zero keys + hit-bitmask (graph-replay safe, deterministic).
// ---------------------------------------------------------------------------
__global__ void sfp_init(unsigned* __restrict__ ws32) {
  int i = blockIdx.x * 256 + threadIdx.x;
  if (i < WS_ZERO_WORDS) ws32[i] = 0u;
}

// ---------------------------------------------------------------------------
// Kernel 1: grid (N_/256, NQ_, B_) = 1280 blocks. Each block async-stages one
// teacher quarter (512 boxes) into LDS, BEV-ifies it, then each thread scans
// its student box against the quarter with a branchless 8-wide margin loop
// (margin = inter - 0.3*union > 0  <=>  IoU > 0.3). Hits -> atomicOr bitmask.
// ---------------------------------------------------------------------------
__global__ void __launch_bounds__(256)
sfp_iou(const float* __restrict__ hi_rois, const float* __restrict__ lo_rois,
        const float* __restrict__ t_rois, unsigned* __restrict__ hitbits) {
  __shared__ float  raw[TQ_ * D_];   // 18 KB async landing zone (16B aligned)
  __shared__ float4 tbox[TQ_];       //  8 KB bev xyxy
  __shared__ float  tarea[TQ_];      //  2 KB bev area

  const int tid = threadIdx.x;
  const int q   = blockIdx.y;
  const int b   = blockIdx.z;
  const int n   = blockIdx.x * 256 + tid;          // grid sized exactly to N_

  const float* srow = (n < H_)
      ? (hi_rois + ((size_t)b * H_ + n) * D_)
      : (lo_rois + ((size_t)b * L_ + (n - H_)) * D_);
  __builtin_prefetch(srow, 0, 1);                  // -> global_prefetch_b8

  // Async-stage the raw teacher quarter: 1152 float4s over 256 lanes.
  const float* src = t_rois + ((size_t)b * T_ + (size_t)q * TQ_) * D_;
  for (int i = tid; i < (TQ_ * D_) / 4; i += 256)
    async_cp16(&raw[i * 4], &src[i * 4]);
  async_wait_all();          // this wave's async writes landed
  __syncthreads();           // all waves' writes visible

  for (int t = tid; t < TQ_; t += 256) {           // stride-9: conflict-free
    const float* r = &raw[t * D_];
    nearest_bev(r[0], r[1], r[3], r[4], r[6], &tbox[t], &tarea[t]);
  }
  __syncthreads();

  float4 a; float sa;
  nearest_bev(srow[0], srow[1], srow[3], srow[4], srow[6], &a, &sa);

  // Branchless 8-wide tiles; uniform LDS addresses -> bank broadcast.
  bool hit = false;
  for (int t0 = 0; t0 < TQ_; t0 += 8) {
    float m = -1.0f;
#pragma unroll
    for (int k = 0; k < 8; ++k) {
      float4 tb   = tbox[t0 + k];
      float iw    = fmaxf(fminf(a.z, tb.z) - fmaxf(a.x, tb.x), 0.0f);
      float ih    = fmaxf(fminf(a.w, tb.w) - fmaxf(a.y, tb.y), 0.0f);
      float inter = iw * ih;
      float uni   = fmaxf(sa + tarea[t0 + k] - inter, 1e-6f);
      m = fmaxf(m, inter - IOU_T * uni);           // >0 iff IoU > 0.3
    }
    if (m > 0.0f) { hit = true; break; }           // early exit per tile
  }

  if (hit) {
    unsigned idx = (unsigned)(b * N_ + n);
    atomicOr(&hitbits[idx >> 5], 1u << (idx & 31));
  }
}

// ---------------------------------------------------------------------------
// Kernel 2: students with no hit in any quarter are "specific"; fold
// (score,n) into the per-(b,class) atomicMax key. Grid: B_*N_/256 = 320.
// ---------------------------------------------------------------------------
__global__ void __launch_bounds__(256)
sfp_select(const float* __restrict__ hi_sc, const int* __restrict__ hi_lb,
           const float* __restrict__ lo_sc, const int* __restrict__ lo_lb,
           const unsigned* __restrict__ hitbits,
           unsigned long long* __restrict__ keys) {
  int idx = blockIdx.x * 256 + threadIdx.x;        // = b*N_ + n, exact grid
  int b = idx / N_, n = idx % N_;
  bool hit = (hitbits[idx >> 5] >> (idx & 31)) & 1u;
  if (!hit) {
    float sc = (n < H_) ? hi_sc[b * H_ + n] : lo_sc[b * L_ + (n - H_)];
    int   lb = (n < H_) ? hi_lb[b * H_ + n] : lo_lb[b * L_ + (n - H_)];
    if (lb >= 0 && lb < C_) {
      // scores in [0,1): bit pattern order-preserving; +1 marks "valid".
      unsigned long long key =
          ((unsigned long long)(__float_as_uint(sc) + 1u) << 32) | (unsigned)n;
      atomicMax(&keys[b * C_ + lb], key);
    }
  }
}

// ---------------------------------------------------------------------------
// Kernel 3: decode 160 keys -> outputs. Invalid class mirrors the reference:
// argmax over all -1e30 picks index 0 -> rois[b,0]/scores[b,0], label=-1.
// Output layout (float): rois[1440] | scores[160] | labels[160] | valid[160].
// ---------------------------------------------------------------------------
__global__ void sfp_final(const float* __restrict__ hi_rois,
                          const float* __restrict__ hi_sc,
                          const float* __restrict__ lo_rois,
                          const unsigned long long* __restrict__ keys,
                          float* __restrict__ out) {
  int t = blockIdx.x * blockDim.x + threadIdx.x;
  if (t >= B_ * C_) return;
  int b = t / C_, c = t % C_;

  float* o_rois = out;
  float* o_sc   = out + (size_t)B_ * C_ * D_;
  float* o_lb   = o_sc + B_ * C_;
  float* o_vd   = o_lb + B_ * C_;

  unsigned long long key = keys[t];
  int n; float sc, lbf, vd;
  if (key != 0ull) {
    n   = (int)(unsigned)key;
    sc  = __uint_as_float((unsigned)(key >> 32) - 1u);
    lbf = (float)c; vd = 1.0f;
  } else {
    n   = 0;
    sc  = hi_sc[b * H_ + 0];
    lbf = -1.0f; vd = 0.0f;
  }
  const float* srow = (n < H_)
      ? (hi_rois + ((size_t)b * H_ + n) * D_)
      : (lo_rois + ((size_t)b * L_ + (n - H_)) * D_);
  for (int k = 0; k < D_; ++k) o_rois[t * D_ + k] = srow[k];
  o_sc[t] = sc; o_lb[t] = lbf; o_vd[t] = vd;
}

// ---------------------------------------------------------------------------
extern "C" void kernel_launch(void* const* d_in, const int* in_sizes, int n_in,
                              void* d_out, int out_size, void* d_ws,
                              size_t ws_size, hipStream_t stream) {
  const float* hi_rois = (const float*)d_in[0];
  const float* hi_sc   = (const float*)d_in[1];
  const int*   hi_lb   = (const int*)  d_in[2];
  const float* lo_rois = (const float*)d_in[3];
  const float* lo_sc   = (const float*)d_in[4];
  const int*   lo_lb   = (const int*)  d_in[5];
  const float* t_rois  = (const float*)d_in[6];

  unsigned long long* keys = (unsigned long long*)d_ws;        // 1280 B
  unsigned* hitbits = (unsigned*)((char*)d_ws + B_ * C_ * 8);  // 10 KB
  float* out = (float*)d_out;

  sfp_init<<<(WS_ZERO_WORDS + 255) / 256, 256, 0, stream>>>((unsigned*)d_ws);
  sfp_iou<<<dim3(N_ / 256, NQ_, B_), 256, 0, stream>>>(
      hi_rois, lo_rois, t_rois, hitbits);
  sfp_select<<<(B_ * N_) / 256, 256, 0, stream>>>(
      hi_sc, hi_lb, lo_sc, lo_lb, hitbits, keys);
  sfp_final<<<1, 256, 0, stream>>>(hi_rois, hi_sc, lo_rois, keys, out);
}